// Pytorch3dRasterzier_65996467470980
// MI455X (gfx1250) — compile-verified
//
#include <hip/hip_runtime.h>
#include <hip/hip_bf16.h>

// ---------------------------------------------------------------------------
// PyTorch3D-style barycentric interpolation (N=8, H=W=1024, K=1, F=20908, D=3)
//   out[n,d,h,w] = mask * sum_v bary[n,h,w,0,v] * attrs[face, v, d]
// Memory-bound (AI ~0.3 FLOP/B). Strategy:
//   * gfx1250 async engine (global_load_async_to_lds_b128, ASYNCcnt) double-
//     buffers the streaming operands (face ids + barys) tile-by-tile into LDS,
//     overlapping ingest with the latency-dominant random attribute gather.
//   * NT cache hints on the stream-once traffic (staging loads + output
//     stores) so the 6 MB attribute table stays resident in the 192 MB L2.
//   * B128 vector loads/stores throughout; channel-plane stores coalesced.
// ---------------------------------------------------------------------------

constexpr int      THREADS         = 256;           // 8 wave32
constexpr int      TILE_PX         = 1024;          // pixels per tile
constexpr int      TILES_PER_BLOCK = 4;
constexpr int      BLOCK_PX        = TILE_PX * TILES_PER_BLOCK;   // 4096
constexpr unsigned HW              = 1u << 20;      // H*W
constexpr int      HW_SHIFT        = 20;

typedef float v4f __attribute__((ext_vector_type(4)));

// ---- CDNA5 async global->LDS copy (16B per lane), tracked by ASYNCcnt ------
// th:TH_LOAD_NT: streamed exactly once -> don't let it occupy L2.
__device__ __forceinline__ void async_copy_b128(unsigned lds_off, const void* gaddr) {
  asm volatile("global_load_async_to_lds_b128 %0, %1, off th:TH_LOAD_NT"
               :: "v"(lds_off), "v"(gaddr)
               : "memory");
}

#if defined(__has_builtin)
#if __has_builtin(__builtin_amdgcn_s_wait_asynccnt)
#define WAIT_ASYNC(n) __builtin_amdgcn_s_wait_asynccnt(n)
#endif
#endif
#ifndef WAIT_ASYNC
#define WAIT_ASYNC(n) asm volatile("s_wait_asynccnt " #n ::: "memory")
#endif

__device__ __forceinline__ unsigned lds_offset(const void* p) {
  // generic (flat) shared address: low 32 bits == workgroup-relative LDS offset
  return (unsigned)(size_t)p;
}

__global__ __launch_bounds__(THREADS)
void rast_interp_kernel(const int*   __restrict__ p2f,
                        const float* __restrict__ bary,
                        const float* __restrict__ attrs,
                        float*       __restrict__ out)
{
  __shared__ int   s_idx[2][TILE_PX];        //  2 * 4 KB
  __shared__ float s_bar[2][TILE_PX * 3];    //  2 * 12 KB

  const int    t    = threadIdx.x;
  const size_t base = (size_t)blockIdx.x * BLOCK_PX;

  // Stage one tile (face ids + barys) into LDS buffer `buf` via async engine.
  auto stage = [&](int buf, size_t pbase) {
    // 1024 ints = 4 KB : one b128 per thread
    async_copy_b128(lds_offset(&s_idx[buf][t * 4]), p2f + pbase + (size_t)t * 4);
    // 3072 floats = 12 KB : three b128 per thread (48B, 16B aligned)
    const float* bs = bary + pbase * 3 + (size_t)t * 12;
    const unsigned bo = lds_offset(&s_bar[buf][t * 12]);
    async_copy_b128(bo +  0, bs + 0);
    async_copy_b128(bo + 16, bs + 4);
    async_copy_b128(bo + 32, bs + 8);
  };

  stage(0, base);                                   // prologue: tile 0

  for (int tile = 0; tile < TILES_PER_BLOCK; ++tile) {
    const int buf = tile & 1;

    if (tile + 1 < TILES_PER_BLOCK) {
      stage(buf ^ 1, base + (size_t)(tile + 1) * TILE_PX);  // overlap next ingest
      WAIT_ASYNC(4);   // async loads retire in order: <=4 outstanding => current tile landed
    } else {
      WAIT_ASYNC(0);
    }
    __syncthreads();   // all waves' staged data visible

    // ---- compute 4 pixels per thread -------------------------------------
    const size_t p0 = base + (size_t)tile * TILE_PX + (size_t)t * 4;

    const int4   f4 = *reinterpret_cast<const int4*>(&s_idx[buf][t * 4]);
    const float* wb = &s_bar[buf][t * 12];
    const float4 wv0 = *reinterpret_cast<const float4*>(wb + 0);
    const float4 wv1 = *reinterpret_cast<const float4*>(wb + 4);
    const float4 wv2 = *reinterpret_cast<const float4*>(wb + 8);

    const float w[12]  = {wv0.x, wv0.y, wv0.z, wv0.w,
                          wv1.x, wv1.y, wv1.z, wv1.w,
                          wv2.x, wv2.y, wv2.z, wv2.w};
    const int   fi4[4] = {f4.x, f4.y, f4.z, f4.w};

    float R[4], G[4], B[4];
#pragma unroll
    for (int i = 0; i < 4; ++i) {
      const int   fi = fi4[i];
      const float m  = (fi >= 0) ? 1.0f : 0.0f;          // background mask
      const size_t f = (size_t)((fi >= 0) ? fi : 0);     // clamp like reference
      const float* A = attrs + f * 9;                    // (3 verts x 3 chan), L2-resident (RT)

      const float b0 = w[i * 3 + 0], b1 = w[i * 3 + 1], b2 = w[i * 3 + 2];
      const float a0 = A[0], a1 = A[1], a2 = A[2];
      const float a3 = A[3], a4 = A[4], a5 = A[5];
      const float a6 = A[6], a7 = A[7], a8 = A[8];

      R[i] = m * fmaf(b2, a6, fmaf(b1, a3, b0 * a0));
      G[i] = m * fmaf(b2, a7, fmaf(b1, a4, b0 * a1));
      B[i] = m * fmaf(b2, a8, fmaf(b1, a5, b0 * a2));
    }

    // ---- coalesced channel-plane NT stores (out is (N,3,H,W)) ------------
    // Output is write-once: TH_STORE_NT keeps it out of L2 so the attribute
    // gather table stays cached.
    const size_t n     = p0 >> HW_SHIFT;
    const size_t hwoff = p0 & (size_t)(HW - 1);
    float* o = out + n * 3 * (size_t)HW + hwoff;

    const v4f rv = {R[0], R[1], R[2], R[3]};
    const v4f gv = {G[0], G[1], G[2], G[3]};
    const v4f bv = {B[0], B[1], B[2], B[3]};
    __builtin_nontemporal_store(rv, reinterpret_cast<v4f*>(o));
    __builtin_nontemporal_store(gv, reinterpret_cast<v4f*>(o + (size_t)HW));
    __builtin_nontemporal_store(bv, reinterpret_cast<v4f*>(o + 2 * (size_t)HW));

    __syncthreads();   // safe to overwrite `buf` with tile+2's async stage
  }
}

extern "C" void kernel_launch(void* const* d_in, const int* in_sizes, int n_in,
                              void* d_out, int out_size, void* d_ws, size_t ws_size,
                              hipStream_t stream) {
  (void)n_in; (void)out_size; (void)d_ws; (void)ws_size;
  const int*   p2f   = (const int*)  d_in[0];   // (N,H,W,K) int32
  const float* bary  = (const float*)d_in[1];   // (N,H,W,K,3) f32
  const float* attrs = (const float*)d_in[2];   // (N,F,3,D) f32
  float*       out   = (float*)d_out;           // (N,3,H,W) f32

  const int total_px = in_sizes[0];             // N*H*W*K = 8388608
  const int blocks   = total_px / BLOCK_PX;     // 2048

  rast_interp_kernel<<<dim3(blocks), dim3(THREADS), 0, stream>>>(p2f, bary, attrs, out);
}